// conGCN_76012331205189
// MI455X (gfx1250) — compile-verified
//
#include <hip/hip_runtime.h>
#include <hip/hip_bf16.h>
#include <utility>

typedef float v2f __attribute__((ext_vector_type(2)));
typedef float v8f __attribute__((ext_vector_type(8)));

#define NHID 128
#define BN_EPS 1e-5f
#define GAMMA 0.01f

// ---------------------------------------------------------------------------
// fp32 WMMA GEMM, compile-time K, Nc, TN (16-col tiles per wave).
// C[M,Nc] = (acc ? C : 0) + A[M,K] @ W[K,Nc]
// - Each wave computes a 16 x (16*TN) strip (TN accumulators, shared A frag).
// - K/Nc compile-time  -> all loads are constant-immediate off lane-fixed bases.
// - 2-deep software pipeline: fragments for k+4 are fetched before the k-step
//   WMMAs issue, so loads overlap matrix math instead of s_wait_loadcnt 0.
// ---------------------------------------------------------------------------
template <int K, int Nc, int TN>
__global__ void gemm_wmma_tpl(const float* __restrict__ A,
                              const float* __restrict__ W,
                              float* __restrict__ C,
                              int M, int accumulate) {
  int gwave = (int)((blockIdx.x * blockDim.x + threadIdx.x) >> 5);
  int lane  = threadIdx.x & 31;
  const int strips = Nc / (16 * TN);
  int tilesM = (M + 15) >> 4;
  if (gwave >= tilesM * strips) return;          // wave-uniform exit
  int tm = gwave / strips;
  int ts = gwave - tm * strips;
  int m0 = tm << 4;
  int n0 = ts * (16 * TN);
  int l15 = lane & 15;
  int hi  = lane >> 4;                           // 0: K=2k..2k+1  1: K=2k+2..2k+3

  int arow = m0 + l15; if (arow >= M) arow = M - 1;
  const float* __restrict__ Ap = A + (size_t)arow * K + 2 * hi;     // lane-fixed base
  const float* __restrict__ Wp = W + (size_t)(2 * hi) * Nc + n0 + l15;
  float* __restrict__ Cp = C + (size_t)(m0 + (hi << 3)) * Nc + n0 + l15;

  v8f c[TN];
#pragma unroll
  for (int t = 0; t < TN; ++t) {
#pragma unroll
    for (int r = 0; r < 8; ++r) {
      if (accumulate) {
        int rr = m0 + (hi << 3) + r;
        c[t][r] = (rr < M) ? Cp[(size_t)r * Nc + t * 16] : 0.0f;
      } else {
        c[t][r] = 0.0f;
      }
    }
  }

  // pipeline prologue: fragments for k0 = 0
  v2f a0, b0[TN];
  a0.x = Ap[0];
  a0.y = Ap[1];
#pragma unroll
  for (int t = 0; t < TN; ++t) {
    b0[t].x = Wp[t * 16];
    b0[t].y = Wp[Nc + t * 16];
  }

#pragma unroll
  for (int k0 = 0; k0 < K; k0 += 4) {
    v2f a1, b1[TN];
    if (k0 + 4 < K) {                            // prefetch next k-step
      a1.x = Ap[k0 + 4];
      a1.y = Ap[k0 + 5];
#pragma unroll
      for (int t = 0; t < TN; ++t) {
        b1[t].x = Wp[(size_t)(k0 + 4) * Nc + t * 16];
        b1[t].y = Wp[(size_t)(k0 + 5) * Nc + t * 16];
      }
    }
#pragma unroll
    for (int t = 0; t < TN; ++t) {
      c[t] = __builtin_amdgcn_wmma_f32_16x16x4_f32(false, a0, false, b0[t],
                                                   (short)0, c[t], false, false);
    }
    a0 = a1;
#pragma unroll
    for (int t = 0; t < TN; ++t) b0[t] = b1[t];
  }

#pragma unroll
  for (int t = 0; t < TN; ++t) {
#pragma unroll
    for (int r = 0; r < 8; ++r) {
      int rr = m0 + (hi << 3) + r;
      if (rr < M) Cp[(size_t)r * Nc + t * 16] = c[t][r];
    }
  }
}

// ---------------------------------------------------------------------------
// SpMM scatter: out[row[e], :] += coef * w[e] * src[col[e], :]  (F = 128)
// 32 threads per edge, float4 gather, 4 f32 atomics (L2-side).
// ---------------------------------------------------------------------------
__global__ void spmm_kernel(float* __restrict__ out,
                            const int* __restrict__ rows,
                            const int* __restrict__ cols,
                            const float* __restrict__ ew,
                            const float* __restrict__ src,
                            float coef, int E) {
  long tid = (long)blockIdx.x * blockDim.x + threadIdx.x;
  long e = tid >> 5;
  if (e >= E) return;
  int f = (int)(tid & 31) * 4;
  float w = coef * ew[e];
  int r = rows[e], c = cols[e];
  float4 v = *(const float4*)(src + (size_t)c * NHID + f);
  float* o = out + (size_t)r * NHID + f;
  atomicAdd(o + 0, w * v.x);
  atomicAdd(o + 1, w * v.y);
  atomicAdd(o + 2, w * v.z);
  atomicAdd(o + 3, w * v.w);
}

__global__ void fill_kernel(float* __restrict__ p, float v, long n) {
  long i = (long)blockIdx.x * blockDim.x + threadIdx.x;
  if (i < n) p[i] = v;
}

__global__ void edge_deg_kernel(const int* __restrict__ rows,
                                const float* __restrict__ val,
                                float* __restrict__ deg, int E) {
  int e = blockIdx.x * blockDim.x + threadIdx.x;
  if (e < E) atomicAdd(&deg[rows[e]], val[e]);
}

__global__ void rsqrt_kernel(float* __restrict__ dis,
                             const float* __restrict__ deg, int n) {
  int i = blockIdx.x * blockDim.x + threadIdx.x;
  if (i < n) dis[i] = rsqrtf(fmaxf(deg[i], 1e-6f));
}

__global__ void nvals_kernel(float* __restrict__ nv,
                             const int* __restrict__ rows,
                             const int* __restrict__ cols,
                             const float* __restrict__ val,
                             const float* __restrict__ dis, int E) {
  int e = blockIdx.x * blockDim.x + threadIdx.x;
  if (e < E) nv[e] = dis[rows[e]] * val[e] * dis[cols[e]];
}

// p = a*p + b*q  (t2 = 2*lhat(t1) - x)
__global__ void axpby_kernel(float* __restrict__ p, const float* __restrict__ q,
                             float a, float b, long n) {
  long i = (long)blockIdx.x * blockDim.x + threadIdx.x;
  if (i < n) p[i] = a * p[i] + b * q[i];
}

// wadj[k] = cheb_w[k] + GAMMA * ||cheb_w[k]||_F    (one block per k, 3 blocks)
__global__ void chebadj_kernel(const float* __restrict__ cw,
                               float* __restrict__ wadj) {
  int k = blockIdx.x;
  const float* src = cw + (size_t)k * NHID * NHID;
  float* dst = wadj + (size_t)k * NHID * NHID;
  float s = 0.f;
  for (int i = threadIdx.x; i < NHID * NHID; i += blockDim.x) {
    float v = src[i]; s += v * v;
  }
  __shared__ float red[256];
  __shared__ float fro;
  int t = threadIdx.x;
  red[t] = s; __syncthreads();
  for (int st = 128; st > 0; st >>= 1) {
    if (t < st) red[t] += red[t + st];
    __syncthreads();
  }
  if (t == 0) fro = sqrtf(red[0]);
  __syncthreads();
  float f = GAMMA * fro;
  for (int i = threadIdx.x; i < NHID * NHID; i += blockDim.x)
    dst[i] = src[i] + f;
}

// out = sigm(alpha)*gcn + (1-sigm(alpha))*0.001*cheb  (bias dropped: BN cancels it)
__global__ void combine_kernel(float* __restrict__ out,
                               const float* __restrict__ ogcn,
                               const float* __restrict__ ocheb,
                               const float* __restrict__ alpha, long n) {
  long i = (long)blockIdx.x * blockDim.x + threadIdx.x;
  if (i >= n) return;
  float a = 1.f / (1.f + expf(-alpha[0]));
  out[i] = a * ogcn[i] + (1.f - a) * 0.001f * ocheb[i];
}

// per-feature batch mean / rstd (training-mode BN, biased variance)
__global__ void bn_stats_kernel(const float* __restrict__ x,
                                float* __restrict__ mean,
                                float* __restrict__ rstd, int n, int F) {
  int f = blockIdx.x;
  float s = 0.f, sq = 0.f;
  for (int i = threadIdx.x; i < n; i += blockDim.x) {
    float v = x[(size_t)i * F + f];
    s += v; sq += v * v;
  }
  __shared__ float r1[256], r2[256];
  int t = threadIdx.x;
  r1[t] = s; r2[t] = sq; __syncthreads();
  for (int st = 128; st > 0; st >>= 1) {
    if (t < st) { r1[t] += r1[t + st]; r2[t] += r2[t + st]; }
    __syncthreads();
  }
  if (t == 0) {
    float m = r1[0] / (float)n;
    float var = r2[0] / (float)n - m * m;
    if (var < 0.f) var = 0.f;
    mean[f] = m;
    rstd[f] = rsqrtf(var + BN_EPS);
  }
}

__global__ void bn_apply_kernel(float* __restrict__ x,
                                const float* __restrict__ mean,
                                const float* __restrict__ rstd,
                                const float* __restrict__ g,
                                const float* __restrict__ b,
                                long n, int F) {
  long i = (long)blockIdx.x * blockDim.x + threadIdx.x;
  if (i >= n) return;
  int f = (int)(i % F);
  float v = (x[i] - mean[f]) * rstd[f] * g[f] + b[f];
  x[i] = v > 0.f ? v : (expf(v) - 1.f);     // ELU (every BN here is followed by ELU)
}

__global__ void concat_kernel(float* __restrict__ out,
                              const float* __restrict__ xe,
                              const float* __restrict__ xs, long n) {
  long i = (long)blockIdx.x * blockDim.x + threadIdx.x;
  if (i >= n) return;
  long row = i >> 8;           // /256
  int f = (int)(i & 255);
  out[i] = (f < NHID) ? xe[row * NHID + f] : xs[row * NHID + (f - NHID)];
}

__global__ void logsoftmax_kernel(const float* __restrict__ in,
                                  const float* __restrict__ bias,
                                  float* __restrict__ out, int n) {
  int i = blockIdx.x * blockDim.x + threadIdx.x;
  if (i >= n) return;
  float v[16];
  float mx = -3.4e38f;
#pragma unroll
  for (int c = 0; c < 16; ++c) {
    v[c] = in[(size_t)i * 16 + c] + bias[c];
    mx = fmaxf(mx, v[c]);
  }
  float s = 0.f;
#pragma unroll
  for (int c = 0; c < 16; ++c) s += expf(v[c] - mx);
  float ls = logf(s) + mx;
#pragma unroll
  for (int c = 0; c < 16; ++c) out[(size_t)i * 16 + c] = v[c] - ls;
}

// ---------------------------------------------------------------------------
extern "C" void kernel_launch(void* const* d_in, const int* in_sizes, int n_in,
                              void* d_out, int out_size, void* d_ws, size_t ws_size,
                              hipStream_t stream) {
  const float* x  = (const float*)d_in[0];
  const int*   a0 = (const int*)d_in[1];
  const float* v0 = (const float*)d_in[2];
  const int*   a1 = (const int*)d_in[3];
  const float* v1 = (const float*)d_in[4];

  const int N = in_sizes[0] / NHID;
  const int E = in_sizes[2];
  const int* row0 = a0;       const int* col0 = a0 + E;
  const int* row1 = a1;       const int* col1 = a1 + E;

  // ---- flattened params (dict traversal order) ----
  int pi = 5;
  auto nf = [&]() { return (const float*)d_in[pi++]; };
  struct GcnP { const float *w, *cw, *alpha, *b; };
  struct BnP  { const float *g, *b; };
  GcnP gin_exp{nf(), nf(), nf(), nf()};
  GcnP gin_sp {nf(), nf(), nf(), nf()};
  BnP  bn_in_exp{nf(), nf()};
  BnP  bn_in_sp {nf(), nf()};
  GcnP cgc_exp[2], cgc_sp[2];
  for (int i = 0; i < 2; ++i) cgc_exp[i] = GcnP{nf(), nf(), nf(), nf()};
  for (int i = 0; i < 2; ++i) cgc_sp[i]  = GcnP{nf(), nf(), nf(), nf()};
  BnP bn_ce[2], bn_cs[2];
  for (int i = 0; i < 2; ++i) bn_ce[i] = BnP{nf(), nf()};
  for (int i = 0; i < 2; ++i) bn_cs[i] = BnP{nf(), nf()};
  const float* fc11_w = nf(); (void)nf();            // fc_out11.b cancelled by BN
  BnP bn_out1{nf(), nf()};
  const float* fch_w = nf(); (void)nf();             // fc_hid.b cancelled by BN
  BnP bn_hid{nf(), nf()};
  const float* fc12_w = nf();
  const float* fc12_b = nf();

  // ---- workspace layout (floats) ----
  float* ws = (float*)d_ws;
  const size_t NF = (size_t)N * NHID;
  float* B0 = ws;                 // x_exp
  float* B1 = B0 + NF;            // x_sp
  float* B2 = B1 + NF;            // support / final logits
  float* B3 = B2 + NF;            // gcn accumulator
  float* B4 = B3 + NF;            // t1
  float* B5 = B4 + NF;            // t2 (lhat scratch)
  float* B6 = B5 + NF;            // cheb accumulator
  float* B7 = B6 + NF;            // ping-pong spare
  float* B8 = B7 + NF;            // concat [N,256]
  float* B9 = B8 + 2 * NF;        // fc stage [N,128]
  float* deg  = B9 + NF;
  float* dis0 = deg + N;
  float* dis1 = dis0 + N;
  float* nv0  = dis1 + N;
  float* nv1  = nv0 + E;
  float* wadj = nv1 + E;          // 3*128*128
  float* mean = wadj + 3 * NHID * NHID;
  float* rstd = mean + 256;

  auto blocks1 = [](long n) { return (unsigned)((n + 255) / 256); };
  auto fill = [&](float* p, float v, long n) {
    fill_kernel<<<blocks1(n), 256, 0, stream>>>(p, v, n);
  };
  // shape-specialized WMMA GEMM dispatch
  auto gemm = [&](const float* A, const float* Wm, float* C,
                  int M, int K, int Nc, int acc) {
    long tilesM = (M + 15) / 16;
    if (K == 128 && Nc == 128) {           // 2 strips of 16x64 per tile row
      long waves = tilesM * 2;
      gemm_wmma_tpl<128, 128, 4><<<blocks1(waves * 32), 256, 0, stream>>>(A, Wm, C, M, acc);
    } else if (K == 256 && Nc == 128) {
      long waves = tilesM * 2;
      gemm_wmma_tpl<256, 128, 4><<<blocks1(waves * 32), 256, 0, stream>>>(A, Wm, C, M, acc);
    } else {                               // K=128, Nc=16
      long waves = tilesM;
      gemm_wmma_tpl<128, 16, 1><<<blocks1(waves * 32), 256, 0, stream>>>(A, Wm, C, M, acc);
    }
  };
  auto spmm = [&](float* out, const int* rows, const int* cols,
                  const float* ew, const float* src, float coef) {
    spmm_kernel<<<blocks1((long)E * 32), 256, 0, stream>>>(out, rows, cols, ew, src, coef, E);
  };
  auto bn_elu = [&](float* xb, const BnP& bn, int F) {
    bn_stats_kernel<<<F, 256, 0, stream>>>(xb, mean, rstd, N, F);
    bn_apply_kernel<<<blocks1((long)N * F), 256, 0, stream>>>(xb, mean, rstd, bn.g, bn.b, (long)N * F, F);
  };

  auto gcn_layer = [&](const float* xin, float* xout,
                       const int* rows, const int* cols, const float* vals,
                       const float* nv, const GcnP& p, const BnP& bn) {
    gemm(xin, p.w, B2, N, NHID, NHID, 0);                    // support = xin @ W
    fill(B3, 0.f, NF); spmm(B3, rows, cols, vals, B2, 1.f);  // out_gcn
    fill(B4, 0.f, NF); spmm(B4, rows, cols, nv, xin, -1.f);  // t1 = lhat(x)
    fill(B5, 0.f, NF); spmm(B5, rows, cols, nv, B4, -1.f);   // s = lhat(t1)
    axpby_kernel<<<blocks1(NF), 256, 0, stream>>>(B5, xin, 2.f, -1.f, NF); // t2
    chebadj_kernel<<<3, 256, 0, stream>>>(p.cw, wadj);
    gemm(xin, wadj,                  B6, N, NHID, NHID, 0);  // cheb k=0
    gemm(B4,  wadj + NHID * NHID,    B6, N, NHID, NHID, 1);  // cheb k=1
    gemm(B5,  wadj + 2 * NHID*NHID,  B6, N, NHID, NHID, 1);  // cheb k=2
    combine_kernel<<<blocks1(NF), 256, 0, stream>>>(xout, B3, B6, p.alpha, NF);
    bn_elu(xout, bn, NHID);
  };

  // ---- graph normalization (once per graph) ----
  fill(deg, 0.f, N);
  edge_deg_kernel<<<blocks1(E), 256, 0, stream>>>(row0, v0, deg, E);
  rsqrt_kernel<<<blocks1(N), 256, 0, stream>>>(dis0, deg, N);
  nvals_kernel<<<blocks1(E), 256, 0, stream>>>(nv0, row0, col0, v0, dis0, E);
  fill(deg, 0.f, N);
  edge_deg_kernel<<<blocks1(E), 256, 0, stream>>>(row1, v1, deg, E);
  rsqrt_kernel<<<blocks1(N), 256, 0, stream>>>(dis1, deg, N);
  nvals_kernel<<<blocks1(E), 256, 0, stream>>>(nv1, row1, col1, v1, dis1, E);

  // ---- GCN stack ----
  gcn_layer(x, B0, row0, col0, v0, nv0, gin_exp, bn_in_exp);
  gcn_layer(x, B1, row1, col1, v1, nv1, gin_sp,  bn_in_sp);
  float* xe = B0; float* xs = B1; float* spare = B7;
  for (int i = 0; i < 2; ++i) {
    gcn_layer(xe, spare, row0, col0, v0, nv0, cgc_exp[i], bn_ce[i]);
    std::swap(xe, spare);
    gcn_layer(xs, spare, row1, col1, v1, nv1, cgc_sp[i], bn_cs[i]);
    std::swap(xs, spare);
  }

  // ---- FC head ----
  concat_kernel<<<blocks1((long)N * 256), 256, 0, stream>>>(B8, xe, xs, (long)N * 256);
  gemm(B8, fc11_w, B9, N, 2 * NHID, NHID, 0);
  bn_elu(B9, bn_out1, NHID);
  gemm(B9, fch_w, spare, N, NHID, NHID, 0);
  bn_elu(spare, bn_hid, NHID);
  gemm(spare, fc12_w, B2, N, NHID, 16, 0);
  logsoftmax_kernel<<<blocks1(N), 256, 0, stream>>>(B2, fc12_b, (float*)d_out, N);
}